// SSDAttention_26173530701941
// MI455X (gfx1250) — compile-verified
//
#include <hip/hip_runtime.h>
#include <math.h>

// ---------------- problem constants (match reference) ----------------
#define BSZ      8
#define SEQ_LEN  4096
#define D_IN     1024
#define STATE_N  1024
#define LN_EPS   1e-5f

#define CHUNK    128
#define NCHUNK   (SEQ_LEN / CHUNK)   // 32

// ---------------- WMMA vector types (wave32, gfx1250) ----------------
typedef float v2f __attribute__((ext_vector_type(2)));
typedef float v8f __attribute__((ext_vector_type(8)));

// ---------------- CDNA5 async global->LDS helpers ---------------------
// Each lane copies 16B directly from global memory into LDS (no VGPR
// round trip); tracked by ASYNCcnt.  LDS byte offset = low 32 bits of the
// generic pointer to a __shared__ object (flat aperture keeps the AS(3)
// offset in addr[31:0]).
__device__ __forceinline__ void async_copy16(unsigned lds_off, const float* gptr) {
  asm volatile("global_load_async_to_lds_b128 %0, %1, off"
               :: "v"(lds_off), "v"(gptr)
               : "memory");
}
__device__ __forceinline__ void wait_async0() {
  asm volatile("s_wait_asynccnt 0x0" ::: "memory");
}

// ---------------- GEMM: C[m,n] = sum_k A[m,k] * W[n,k] + bias[n] ------
// 128x128 tile per 256-thread block; 8 wave32 waves, each -> 64(M) x 32(N).
// K blocked by 16 through double-buffered LDS filled with async copies;
// compute path is V_WMMA_F32_16X16X4_F32 (full fp32 precision).
#define TILE_M 128
#define TILE_N 128
#define TILE_K 16
#define LDSS   (TILE_K + 4)   // 20 floats: 16B-aligned float4 slots, low bank conflict

__launch_bounds__(256)
__global__ void gemm_bias_kernel(const float* __restrict__ A,    // M x K
                                 const float* __restrict__ W,    // N x K
                                 const float* __restrict__ bias, // N
                                 float* __restrict__ C,          // M x N
                                 int M, int N, int K) {
  __shared__ float Asl[2][TILE_M * LDSS];
  __shared__ float Bsl[2][TILE_N * LDSS];

  const int tid  = threadIdx.x;
  const int lane = tid & 31;
  const int wave = tid >> 5;
  const int lq   = lane & 15;   // m (A), n (B/C) within 16x16 subtile
  const int lh   = lane >> 4;   // lane half selects K pair / row half

  const int m0 = blockIdx.y * TILE_M;
  const int n0 = blockIdx.x * TILE_N;
  const int wm = (wave & 1) * 64;   // 2 waves along M
  const int wn = (wave >> 1) * 32;  // 4 waves along N

  // Per-thread fill slots: tile = 128x16 floats = 512 float4s; 2 per thread.
  const int j0 = tid;        // float4 index 0..255
  const int j1 = tid + 256;  // float4 index 256..511
  const int row0 = j0 >> 2, kc0 = (j0 & 3) << 2;
  const int row1 = j1 >> 2, kc1 = (j1 & 3) << 2;

  auto issue_tile = [&](int kbase, int buf) {
    async_copy16((unsigned)(size_t)&Asl[buf][row0 * LDSS + kc0],
                 &A[(size_t)(m0 + row0) * K + kbase + kc0]);
    async_copy16((unsigned)(size_t)&Asl[buf][row1 * LDSS + kc1],
                 &A[(size_t)(m0 + row1) * K + kbase + kc1]);
    async_copy16((unsigned)(size_t)&Bsl[buf][row0 * LDSS + kc0],
                 &W[(size_t)(n0 + row0) * K + kbase + kc0]);
    async_copy16((unsigned)(size_t)&Bsl[buf][row1 * LDSS + kc1],
                 &W[(size_t)(n0 + row1) * K + kbase + kc1]);
  };

  v8f acc[4][2] = {};

  // Prologue: fill buffer 0.
  issue_tile(0, 0);
  wait_async0();
  __syncthreads();

  for (int k0 = 0; k0 < K; k0 += TILE_K) {
    const int cur = (k0 / TILE_K) & 1;

    // Overlap: start filling the other buffer while computing on `cur`.
    if (k0 + TILE_K < K) issue_tile(k0 + TILE_K, cur ^ 1);

    const float* __restrict__ As = &Asl[cur][0];
    const float* __restrict__ Bs = &Bsl[cur][0];
#pragma unroll
    for (int kk = 0; kk < TILE_K; kk += 4) {
      // ISA frag layout: VGPR i, lane l -> k = i + 2*(l>>4); m/n = l&15
      const int kf = kk + 2 * lh;
      v2f afrag[4], bfrag[2];
#pragma unroll
      for (int i = 0; i < 4; ++i)
        afrag[i] = *(const v2f*)(&As[(wm + i * 16 + lq) * LDSS + kf]);
#pragma unroll
      for (int j = 0; j < 2; ++j)
        bfrag[j] = *(const v2f*)(&Bs[(wn + j * 16 + lq) * LDSS + kf]);
#pragma unroll
      for (int i = 0; i < 4; ++i)
#pragma unroll
        for (int j = 0; j < 2; ++j)
          acc[i][j] = __builtin_amdgcn_wmma_f32_16x16x4_f32(
              false, afrag[i], false, bfrag[j], (short)0, acc[i][j],
              false, false);
    }

    // My async fills landed; barrier publishes them (and retires reads of
    // `cur` so the next iteration may overwrite it).
    wait_async0();
    __syncthreads();
  }

  // Epilogue: C/D layout VGPR g, lane l -> m = g + 8*(l>>4), n = l&15.
#pragma unroll
  for (int j = 0; j < 2; ++j) {
    const int n  = n0 + wn + j * 16 + lq;
    const float bv = bias[n];
#pragma unroll
    for (int i = 0; i < 4; ++i) {
#pragma unroll
      for (int g = 0; g < 8; ++g) {
        const int m = m0 + wm + i * 16 + g + 8 * lh;
        C[(size_t)m * N + n] = acc[i][j][g] + bv;
      }
    }
  }
}

// ---------------- chunked diagonal linear scan ------------------------
__device__ __forceinline__ float softplus_f(float x) {
  return (x > 20.f) ? x : log1pf(expf(x));
}

// Pass 1: local scan per chunk starting from state 0; record chunk-end state.
__launch_bounds__(256)
__global__ void scan_chunk_local(const float* __restrict__ h,
                                 const float* __restrict__ log_A,
                                 const float* __restrict__ B_p,
                                 float* __restrict__ E) {
  const int idx  = blockIdx.x * 256 + threadIdx.x;  // BSZ*NCHUNK*STATE_N
  const int e    = idx % STATE_N;                   // channel fastest -> coalesced
  const int rest = idx / STATE_N;
  const int c    = rest % NCHUNK;
  const int b    = rest / NCHUNK;

  const float decay = expf(-softplus_f(log_A[e]));
  const float bp    = B_p[e];
  size_t p = ((size_t)b * SEQ_LEN + (size_t)c * CHUNK) * STATE_N + e;
  float s = 0.f;
#pragma unroll 4
  for (int i = 0; i < CHUNK; ++i) {
    s = fmaf(s, decay, bp * h[p]);
    p += STATE_N;
  }
  E[((size_t)b * NCHUNK + c) * STATE_N + e] = s;
}

// Pass 2: serial prefix over the 32 chunk-end states per (b, channel).
__launch_bounds__(256)
__global__ void scan_chunk_carry(const float* __restrict__ E,
                                 const float* __restrict__ log_A,
                                 float* __restrict__ Carry) {
  const int idx = blockIdx.x * 256 + threadIdx.x;   // BSZ*STATE_N
  const int e   = idx % STATE_N;
  const int b   = idx / STATE_N;
  const float dC = expf(-softplus_f(log_A[e]) * (float)CHUNK);  // decay^CHUNK
  float carry = 0.f;
  for (int c = 0; c < NCHUNK; ++c) {
    const size_t o = ((size_t)b * NCHUNK + c) * STATE_N + e;
    Carry[o] = carry;
    carry = fmaf(carry, dC, E[o]);
  }
}

// Pass 3: re-scan with carried initial state, emit y = C_p*s in place over h.
__launch_bounds__(256)
__global__ void scan_chunk_apply(float* __restrict__ hy,
                                 const float* __restrict__ log_A,
                                 const float* __restrict__ B_p,
                                 const float* __restrict__ C_p,
                                 const float* __restrict__ Carry) {
  const int idx  = blockIdx.x * 256 + threadIdx.x;
  const int e    = idx % STATE_N;
  const int rest = idx / STATE_N;
  const int c    = rest % NCHUNK;
  const int b    = rest / NCHUNK;

  const float decay = expf(-softplus_f(log_A[e]));
  const float bp    = B_p[e];
  const float cp    = C_p[e];
  float s = Carry[((size_t)b * NCHUNK + c) * STATE_N + e];
  size_t p = ((size_t)b * SEQ_LEN + (size_t)c * CHUNK) * STATE_N + e;
#pragma unroll 4
  for (int i = 0; i < CHUNK; ++i) {
    s = fmaf(s, decay, bp * hy[p]);
    hy[p] = cp * s;   // each thread touches only its own (b,t,e): safe in place
    p += STATE_N;
  }
}

// ---------------- LayerNorm (in place, one block per row) -------------
__launch_bounds__(256)
__global__ void layernorm_kernel(float* __restrict__ out,
                                 const float* __restrict__ gamma,
                                 const float* __restrict__ beta) {
  __shared__ float ssum[256];
  __shared__ float ssq[256];
  const int tid = threadIdx.x;
  float* row = out + (size_t)blockIdx.x * STATE_N;

  float v[4];
  float s = 0.f, sq = 0.f;
#pragma unroll
  for (int j = 0; j < 4; ++j) {
    v[j] = row[tid + 256 * j];
    s  += v[j];
    sq += v[j] * v[j];
  }
  ssum[tid] = s;
  ssq[tid]  = sq;
  __syncthreads();
  for (int off = 128; off > 0; off >>= 1) {
    if (tid < off) {
      ssum[tid] += ssum[tid + off];
      ssq[tid]  += ssq[tid + off];
    }
    __syncthreads();
  }
  const float mean = ssum[0] * (1.f / STATE_N);
  const float var  = ssq[0] * (1.f / STATE_N) - mean * mean;  // population var
  const float rstd = rsqrtf(var + LN_EPS);
#pragma unroll
  for (int j = 0; j < 4; ++j) {
    const int i = tid + 256 * j;
    row[i] = (v[j] - mean) * rstd * gamma[i] + beta[i];
  }
}

// ---------------- driver ----------------------------------------------
extern "C" void kernel_launch(void* const* d_in, const int* in_sizes, int n_in,
                              void* d_out, int out_size, void* d_ws, size_t ws_size,
                              hipStream_t stream) {
  (void)in_sizes; (void)n_in; (void)out_size; (void)ws_size;

  const float* x     = (const float*)d_in[0];
  const float* W_in  = (const float*)d_in[1];
  const float* b_in  = (const float*)d_in[2];
  const float* log_A = (const float*)d_in[3];
  const float* B_p   = (const float*)d_in[4];
  const float* C_p   = (const float*)d_in[5];
  const float* W_out = (const float*)d_in[6];
  const float* b_out = (const float*)d_in[7];
  const float* gamma = (const float*)d_in[8];
  const float* beta  = (const float*)d_in[9];
  float* out = (float*)d_out;

  const int M = BSZ * SEQ_LEN;     // 32768
  const int N = STATE_N;           // 1024
  const int K = D_IN;              // 1024

  // workspace: h/y buffer (128 MB) + chunk-end states + carries (1 MB each)
  float* h     = (float*)d_ws;
  float* E     = h + (size_t)M * STATE_N;
  float* Carry = E + (size_t)BSZ * NCHUNK * STATE_N;

  dim3 gemm_grid(N / TILE_N, M / TILE_M);   // (8, 256); x-fastest keeps A band in L2
  gemm_bias_kernel<<<gemm_grid, 256, 0, stream>>>(x, W_in, b_in, h, M, N, K);

  const int scan_elems = BSZ * NCHUNK * STATE_N;  // 262144
  scan_chunk_local<<<scan_elems / 256, 256, 0, stream>>>(h, log_A, B_p, E);
  scan_chunk_carry<<<(BSZ * STATE_N) / 256, 256, 0, stream>>>(E, log_A, Carry);
  scan_chunk_apply<<<scan_elems / 256, 256, 0, stream>>>(h, log_A, B_p, C_p, Carry);

  gemm_bias_kernel<<<gemm_grid, 256, 0, stream>>>(h, W_out, b_out, out, M, STATE_N, STATE_N);
  layernorm_kernel<<<M, 256, 0, stream>>>(out, gamma, beta);
}